// Criterion_59493886984529
// MI455X (gfx1250) — compile-verified
//
#include <hip/hip_runtime.h>
#include <stdint.h>

#define HH 512
#define WW 512
#define NB 32
#define NELEM (HH*WW)            // 262144 per problem
#define NPROB (NB*2)             // 64 (sample, channel) problems
#define NCHUNK 32                // blocks per problem
#define CHUNK (NELEM/NCHUNK)     // 8192 elements per block
#define TPB 256
#define V4PT (CHUNK/(TPB*4))     // 8 float4 per thread

// workspace layout in u32 units
#define WS_POSCNT  0u
#define WS_K       (WS_POSCNT + NPROB)
#define WS_CNTGT   (WS_K + NPROB)
#define WS_PREFIX  (WS_CNTGT + NPROB)
#define WS_POSSUM  (WS_PREFIX + NPROB)                 // f32
#define WS_POSPART (WS_POSSUM + NPROB)                 // f32 [NPROB][NCHUNK]
#define WS_SUMPART (WS_POSPART + NPROB*NCHUNK)         // f32 [NPROB][NCHUNK]
#define WS_HIST    (WS_SUMPART + NPROB*NCHUNK)         // u32 [4][NPROB][256]
#define WS_TOTAL   (WS_HIST + 4u*NPROB*256u)           // ~70K u32 (~274 KB)

typedef __attribute__((ext_vector_type(2))) float v2f;
typedef __attribute__((ext_vector_type(8))) float v8f;

__device__ __forceinline__ uint32_t umin_(uint32_t a, uint32_t b) { return a < b ? a : b; }
__device__ __forceinline__ uint32_t umax_(uint32_t a, uint32_t b) { return a > b ? a : b; }
__device__ __forceinline__ float readlane_f(float x, int l) {
  return __uint_as_float(__builtin_amdgcn_readlane(__float_as_uint(x), l));
}

__global__ void zero_ws_kernel(uint32_t* __restrict__ ws, int n) {
  int i = blockIdx.x * blockDim.x + threadIdx.x;
  if (i < n) ws[i] = 0u;
}

// Pass 0: per-block pos count/sum + histogram of top byte of negative loss bits.
__global__ void stats_kernel(const float* __restrict__ pred,
                             const float* __restrict__ region,
                             const float* __restrict__ affinity,
                             uint32_t* __restrict__ ws) {
  __shared__ uint32_t hist[8][256];
  __shared__ float    redf[TPB];
  __shared__ uint32_t redc[TPB];
  const int prob  = blockIdx.x / NCHUNK;
  const int chunk = blockIdx.x % NCHUNK;
  const int b = prob >> 1, c = prob & 1;
  const float* lab = (c == 0 ? region : affinity) + (size_t)b * NELEM;
  const float* pp  = pred + (size_t)prob * NELEM;
  const int base = chunk * CHUNK;
  const int wave = threadIdx.x >> 5;

  for (int i = threadIdx.x; i < 8 * 256; i += TPB) (&hist[0][0])[i] = 0u;
  __syncthreads();

  const float4* p4 = (const float4*)(pp + base);
  const float4* l4 = (const float4*)(lab + base);
  uint32_t pcnt = 0; float psum = 0.0f;
  for (int it = 0; it < V4PT; ++it) {
    int idx = it * TPB + threadIdx.x;
    __builtin_prefetch(&p4[idx + 2 * TPB], 0, 0);
    __builtin_prefetch(&l4[idx + 2 * TPB], 0, 0);
    float4 pv = p4[idx];
    float4 lv = l4[idx];
    float pe[4] = {pv.x, pv.y, pv.z, pv.w};
    float le[4] = {lv.x, lv.y, lv.z, lv.w};
#pragma unroll
    for (int e = 0; e < 4; ++e) {
      float d = pe[e] - le[e];
      float loss = d * d;
      if (le[e] >= 0.1f) { pcnt++; psum += loss; }
      else {
        uint32_t bits = __float_as_uint(loss);
        atomicAdd(&hist[wave][bits >> 24], 1u);
      }
    }
  }
  __syncthreads();
  // fold per-wave histograms into the global per-problem histogram (pass 0)
  for (int bin = threadIdx.x; bin < 256; bin += TPB) {
    uint32_t s = 0;
#pragma unroll
    for (int w = 0; w < 8; ++w) s += hist[w][bin];
    if (s) atomicAdd(&ws[WS_HIST + (uint32_t)prob * 256u + bin], s);
  }
  // deterministic block reduce of pos stats
  redf[threadIdx.x] = psum; redc[threadIdx.x] = pcnt;
  __syncthreads();
  for (int s = TPB / 2; s > 0; s >>= 1) {
    if (threadIdx.x < (unsigned)s) { redf[threadIdx.x] += redf[threadIdx.x + s]; redc[threadIdx.x] += redc[threadIdx.x + s]; }
    __syncthreads();
  }
  if (threadIdx.x == 0) {
    atomicAdd(&ws[WS_POSCNT + prob], redc[0]);
    ((float*)ws)[WS_POSPART + prob * NCHUNK + chunk] = redf[0];
  }
}

// Refinement passes pa = 1..3: histogram next byte among negatives matching prefix.
__global__ void hist_pass_kernel(const float* __restrict__ pred,
                                 const float* __restrict__ region,
                                 const float* __restrict__ affinity,
                                 uint32_t* __restrict__ ws, int pa) {
  __shared__ uint32_t hist[8][256];
  const int prob  = blockIdx.x / NCHUNK;
  const int chunk = blockIdx.x % NCHUNK;
  const int b = prob >> 1, c = prob & 1;
  const float* lab = (c == 0 ? region : affinity) + (size_t)b * NELEM;
  const float* pp  = pred + (size_t)prob * NELEM;
  const int base = chunk * CHUNK;
  const int wave = threadIdx.x >> 5;
  const uint32_t prefix = ws[WS_PREFIX + prob];
  const uint32_t mask = 0xFFFFFFFFu << (32 - 8 * pa);
  const int shift = 24 - 8 * pa;

  for (int i = threadIdx.x; i < 8 * 256; i += TPB) (&hist[0][0])[i] = 0u;
  __syncthreads();

  const float4* p4 = (const float4*)(pp + base);
  const float4* l4 = (const float4*)(lab + base);
  for (int it = 0; it < V4PT; ++it) {
    int idx = it * TPB + threadIdx.x;
    __builtin_prefetch(&p4[idx + 2 * TPB], 0, 0);
    __builtin_prefetch(&l4[idx + 2 * TPB], 0, 0);
    float4 pv = p4[idx];
    float4 lv = l4[idx];
    float pe[4] = {pv.x, pv.y, pv.z, pv.w};
    float le[4] = {lv.x, lv.y, lv.z, lv.w};
#pragma unroll
    for (int e = 0; e < 4; ++e) {
      if (le[e] < 0.1f) {
        float d = pe[e] - le[e];
        uint32_t bits = __float_as_uint(d * d);
        if ((bits & mask) == prefix)
          atomicAdd(&hist[wave][(bits >> shift) & 0xFF], 1u);
      }
    }
  }
  __syncthreads();
  for (int bin = threadIdx.x; bin < 256; bin += TPB) {
    uint32_t s = 0;
#pragma unroll
    for (int w = 0; w < 8; ++w) s += hist[w][bin];
    if (s) atomicAdd(&ws[WS_HIST + ((uint32_t)pa * NPROB + prob) * 256u + bin], s);
  }
}

// Device-side select: walk histogram from top bin, extend prefix, track count-greater.
__global__ void select_kernel(uint32_t* __restrict__ ws, int pa) {
  int p = threadIdx.x;
  if (p >= NPROB) return;
  float* wsf = (float*)ws;
  if (pa == 0) {
    uint32_t pcnt = ws[WS_POSCNT + p];
    float psum = 0.0f;
    for (int i = 0; i < NCHUNK; ++i) psum += wsf[WS_POSPART + p * NCHUNK + i];
    wsf[WS_POSSUM + p] = psum;
    uint32_t neg = (uint32_t)NELEM - pcnt;
    uint32_t k = (pcnt > 0) ? umin_(3u * pcnt, neg) : 500u;
    ws[WS_K + p] = k;
    ws[WS_CNTGT + p] = 0u;
    ws[WS_PREFIX + p] = 0u;
    if (neg == 0u) return;   // degenerate: handled in final kernel
  }
  uint32_t k   = ws[WS_K + p];
  uint32_t cgt = ws[WS_CNTGT + p];
  uint32_t target = k - cgt;  // >= 1 whenever neg > 0
  const uint32_t* h = &ws[WS_HIST + ((uint32_t)pa * NPROB + p) * 256u];
  uint32_t acc = 0; int sel = 0;
  for (int bin = 255; bin >= 0; --bin) {
    uint32_t cc = h[bin];
    if (acc + cc >= target) { sel = bin; break; }
    acc += cc;
  }
  int shift = 24 - 8 * pa;
  ws[WS_PREFIX + p] |= ((uint32_t)sel) << shift;
  ws[WS_CNTGT + p] = cgt + acc;
}

// Sum of negative losses strictly greater than the selected k-th value v.
__global__ void sumgt_kernel(const float* __restrict__ pred,
                             const float* __restrict__ region,
                             const float* __restrict__ affinity,
                             uint32_t* __restrict__ ws) {
  __shared__ float redf[TPB];
  const int prob  = blockIdx.x / NCHUNK;
  const int chunk = blockIdx.x % NCHUNK;
  const int b = prob >> 1, c = prob & 1;
  const float* lab = (c == 0 ? region : affinity) + (size_t)b * NELEM;
  const float* pp  = pred + (size_t)prob * NELEM;
  const int base = chunk * CHUNK;
  const uint32_t vbits = ws[WS_PREFIX + prob];

  const float4* p4 = (const float4*)(pp + base);
  const float4* l4 = (const float4*)(lab + base);
  float s = 0.0f;
  for (int it = 0; it < V4PT; ++it) {
    int idx = it * TPB + threadIdx.x;
    __builtin_prefetch(&p4[idx + 2 * TPB], 0, 0);
    __builtin_prefetch(&l4[idx + 2 * TPB], 0, 0);
    float4 pv = p4[idx];
    float4 lv = l4[idx];
    float pe[4] = {pv.x, pv.y, pv.z, pv.w};
    float le[4] = {lv.x, lv.y, lv.z, lv.w};
#pragma unroll
    for (int e = 0; e < 4; ++e) {
      if (le[e] < 0.1f) {
        float d = pe[e] - le[e];
        float loss = d * d;
        if (__float_as_uint(loss) > vbits) s += loss;   // bit order == value order (loss >= 0)
      }
    }
  }
  redf[threadIdx.x] = s;
  __syncthreads();
  for (int t = TPB / 2; t > 0; t >>= 1) {
    if (threadIdx.x < (unsigned)t) redf[threadIdx.x] += redf[threadIdx.x + t];
    __syncthreads();
  }
  if (threadIdx.x == 0)
    ((float*)ws)[WS_SUMPART + prob * NCHUNK + chunk] = redf[0];
}

// One wave (32 lanes): assemble per-problem losses, grand-total via WMMA f32 16x16x4.
__global__ void final_kernel(uint32_t* __restrict__ ws, float* __restrict__ out) {
  const int lane = threadIdx.x;  // 0..31
  float* wsf = (float*)ws;
  float per[2];
#pragma unroll
  for (int j = 0; j < 2; ++j) {
    int p = lane + j * 32;
    uint32_t pcnt = ws[WS_POSCNT + p];
    uint32_t k    = ws[WS_K + p];
    uint32_t cgt  = ws[WS_CNTGT + p];
    uint32_t vb   = ws[WS_PREFIX + p];
    float psum    = wsf[WS_POSSUM + p];
    float sgt = 0.0f;
    for (int i = 0; i < NCHUNK; ++i) sgt += wsf[WS_SUMPART + p * NCHUNK + i];
    uint32_t neg = (uint32_t)NELEM - pcnt;
    float posi = psum / (float)umax_(pcnt, 1u);   // 0 when pcnt==0
    float nega;
    if (neg == 0u) {
      nega = -1.0f;  // reference: csum[0] of all -1 entries
    } else {
      float v = __uint_as_float(vb);
      nega = (sgt + (float)(k - cgt) * v) / (float)umax_(k, 1u);
    }
    per[j] = posi + nega;  // pcnt==0 path: posi==0, k==500 -> top500 mean
  }
  // Grand total of all 64 values via WMMA: A = values, B = ones.
  // D[m][n] = rowsum_m(A); sum over column 0 = sum of ALL A entries,
  // independent of A's lane<->(m,k) mapping (it is a bijection).
  v2f a; a.x = per[0]; a.y = per[1];
  v2f bm; bm.x = 1.0f; bm.y = 1.0f;
  v8f cm = {};
  cm = __builtin_amdgcn_wmma_f32_16x16x4_f32(
      /*neg_a=*/false, a, /*neg_b=*/false, bm,
      /*c_mod=*/(short)0, cm, /*reuse_a=*/false, /*reuse_b=*/false);
  float s = cm[0] + cm[1] + cm[2] + cm[3] + cm[4] + cm[5] + cm[6] + cm[7];
  // column 0: rows 0..7 live in lane 0, rows 8..15 in lane 16
  float tot = readlane_f(s, 0) + readlane_f(s, 16);
  if (lane == 0) out[0] = tot / (float)NB;
}

extern "C" void kernel_launch(void* const* d_in, const int* in_sizes, int n_in,
                              void* d_out, int out_size, void* d_ws, size_t ws_size,
                              hipStream_t stream) {
  (void)in_sizes; (void)n_in; (void)out_size; (void)ws_size;
  const float* pred     = (const float*)d_in[0];
  const float* region   = (const float*)d_in[1];
  const float* affinity = (const float*)d_in[2];
  float* out = (float*)d_out;
  uint32_t* ws = (uint32_t*)d_ws;

  const int zblocks = (int)((WS_TOTAL + TPB - 1) / TPB);
  zero_ws_kernel<<<dim3(zblocks), dim3(TPB), 0, stream>>>(ws, (int)WS_TOTAL);

  dim3 grid(NPROB * NCHUNK);
  stats_kernel<<<grid, dim3(TPB), 0, stream>>>(pred, region, affinity, ws);
  select_kernel<<<dim3(1), dim3(64), 0, stream>>>(ws, 0);
  for (int pa = 1; pa <= 3; ++pa) {
    hist_pass_kernel<<<grid, dim3(TPB), 0, stream>>>(pred, region, affinity, ws, pa);
    select_kernel<<<dim3(1), dim3(64), 0, stream>>>(ws, pa);
  }
  sumgt_kernel<<<grid, dim3(TPB), 0, stream>>>(pred, region, affinity, ws);
  final_kernel<<<dim3(1), dim3(32), 0, stream>>>(ws, out);
}